// SimpleHippocampus_56410100466006
// MI455X (gfx1250) — compile-verified
//
#include <hip/hip_runtime.h>
#include <math.h>

#define EPS 1e-12f

constexpr int D = 512;
constexpr int NROWS = 500000;
constexpr int WAVES_PER_BLOCK = 10;
constexpr int TPB_A = 32 * WAVES_PER_BLOCK;              // 320 threads = 10 waves
constexpr int ROWS_PER_WAVE = 16;
constexpr int ROWS_PER_BLOCK = WAVES_PER_BLOCK * ROWS_PER_WAVE;  // 160
constexpr int BLOCKS_A = NROWS / ROWS_PER_BLOCK;         // 3125 (exact)
constexpr int NB2 = 256;                                 // phase-2 blocks
constexpr int TPB_B = 256;
constexpr int CHUNK = (NROWS + NB2 - 1) / NB2;           // 1954
constexpr int TOPK = 16;
constexpr int NCAND = NB2 * TOPK;                        // 4096

typedef __attribute__((ext_vector_type(2))) float v2f;
typedef __attribute__((ext_vector_type(8))) float v8f;

// ---------------- Phase 0: query inverse norm ----------------
__global__ __launch_bounds__(256) void qnorm_kernel(const float* __restrict__ q,
                                                    float* __restrict__ qfac) {
    __shared__ float red[256];
    int tid = threadIdx.x;
    float s = 0.f;
    for (int i = tid; i < D; i += 256) {
        float v = q[i];
        s += v * v;
    }
    red[tid] = s;
    __syncthreads();
    for (int off = 128; off > 0; off >>= 1) {
        if (tid < off) red[tid] += red[tid + off];
        __syncthreads();
    }
    if (tid == 0) qfac[0] = 1.0f / fmaxf(sqrtf(red[0]), EPS);
}

// ---------------- Phase 1: cosine scores via V_WMMA_F32_16X16X4_F32 ----------------
// Each wave computes 16 rows' dot(m_row, q) with 128 f32 WMMAs (K=512 in steps of 4),
// plus row sum-of-squares on the side. Score = dot * qfac / max(||row||, eps).
__global__ __launch_bounds__(TPB_A) void score_kernel(const float* __restrict__ q,
                                                      const float* __restrict__ mem,
                                                      const float* __restrict__ qfacp,
                                                      float* __restrict__ scores) {
    __shared__ float qs[D];
    __shared__ float st[WAVES_PER_BLOCK][16];

    int tid = threadIdx.x;
    for (int i = tid; i < D; i += TPB_A) qs[i] = q[i];
    __syncthreads();

    float qfac = qfacp[0];
    int wave = tid >> 5;
    int lane = tid & 31;
    int g = lane >> 4;      // 0: K pair {0,1}; 1: K pair {2,3}
    int r15 = lane & 15;    // row within tile (A-matrix M index)

    long row0 = (long)blockIdx.x * ROWS_PER_BLOCK + (long)wave * ROWS_PER_WAVE;
    const float* rowp = mem + (row0 + r15) * (long)D + 2 * g;

    v8f acc = {0.f, 0.f, 0.f, 0.f, 0.f, 0.f, 0.f, 0.f};
    float ss = 0.f;

    for (int k = 0; k < D; k += 4) {
        // A fragment: lane holds m[row][k+2g], m[row][k+2g+1] (nontemporal: 1GB stream)
        v2f a = __builtin_nontemporal_load((const v2f*)(rowp + k));
        // B fragment: query replicated across all 16 columns
        v2f b;
        b.x = qs[k + g];
        b.y = qs[k + 2 + g];
        ss += a.x * a.x + a.y * a.y;
        // D = A(16x4) * B(4x16) + C  -> v_wmma_f32_16x16x4_f32
        acc = __builtin_amdgcn_wmma_f32_16x16x4_f32(false, a, false, b,
                                                    (short)0, acc, false, false);
    }

    // combine sum-of-squares halves: row r15 total = ss(lane) + ss(lane^16)
    float ssTot = ss + __shfl_xor(ss, 16, 32);

    // C layout: VGPR r -> lanes 0-15: C[M=r][N=lane]; lanes 16-31: C[M=8+r][N=lane-16].
    // Every column identical, so lane 0 owns dots of rows 0..7, lane 16 rows 8..15.
    if (lane == 0) {
#pragma unroll
        for (int r = 0; r < 8; ++r) st[wave][r] = acc[r];
    } else if (lane == 16) {
#pragma unroll
        for (int r = 0; r < 8; ++r) st[wave][8 + r] = acc[r];
    }
    __syncthreads();

    if (lane < 16) {
        float dot = st[wave][lane];
        float mn = fmaxf(sqrtf(ssTot), EPS);
        scores[row0 + lane] = dot * qfac / mn;
    }
}

// ---------------- Phase 2: per-block deterministic top-16 ----------------
__global__ __launch_bounds__(TPB_B) void blocktop_kernel(const float* __restrict__ scores,
                                                         float* __restrict__ candS,
                                                         int* __restrict__ candI) {
    __shared__ float sc[CHUNK];
    __shared__ float rs[TPB_B];
    __shared__ int ri[TPB_B];

    int tid = threadIdx.x;
    int base = blockIdx.x * CHUNK;

    for (int i = tid; i < CHUNK; i += TPB_B) {
        int gi = base + i;
        sc[i] = (gi < NROWS) ? scores[gi] : -INFINITY;
    }
    __syncthreads();

    for (int it = 0; it < TOPK; ++it) {
        float best = -INFINITY;
        int bi = 0x7fffffff;
        for (int i = tid; i < CHUNK; i += TPB_B) {
            float v = sc[i];
            if (v > best || (v == best && i < bi)) { best = v; bi = i; }
        }
        rs[tid] = best; ri[tid] = bi;
        __syncthreads();
        for (int off = TPB_B / 2; off > 0; off >>= 1) {
            if (tid < off) {
                float ov = rs[tid + off]; int oi = ri[tid + off];
                if (ov > rs[tid] || (ov == rs[tid] && oi < ri[tid])) {
                    rs[tid] = ov; ri[tid] = oi;
                }
            }
            __syncthreads();
        }
        if (tid == 0) {
            candS[blockIdx.x * TOPK + it] = rs[0];
            candI[blockIdx.x * TOPK + it] = base + ri[0];
            sc[ri[0]] = -INFINITY;
        }
        __syncthreads();
    }
}

// ---------------- Phase 3: merge 4096 candidates, gather rows ----------------
__global__ __launch_bounds__(256) void merge_gather_kernel(const float* __restrict__ candS,
                                                           const int* __restrict__ candI,
                                                           const float* __restrict__ mem,
                                                           float* __restrict__ out) {
    __shared__ float cs[NCAND];
    __shared__ int ci[NCAND];
    __shared__ float rs[256];
    __shared__ int ri[256];
    __shared__ int rslot[256];
    __shared__ float topS[TOPK];
    __shared__ int topIdx[TOPK];

    int tid = threadIdx.x;
    for (int i = tid; i < NCAND; i += 256) {
        cs[i] = candS[i];
        ci[i] = candI[i];
    }
    __syncthreads();

    for (int it = 0; it < TOPK; ++it) {
        float best = -INFINITY;
        int bi = 0x7fffffff;
        int bslot = 0;
        for (int i = tid; i < NCAND; i += 256) {
            float v = cs[i];
            int idx = ci[i];
            if (v > best || (v == best && idx < bi)) { best = v; bi = idx; bslot = i; }
        }
        rs[tid] = best; ri[tid] = bi; rslot[tid] = bslot;
        __syncthreads();
        for (int off = 128; off > 0; off >>= 1) {
            if (tid < off) {
                if (rs[tid + off] > rs[tid] ||
                    (rs[tid + off] == rs[tid] && ri[tid + off] < ri[tid])) {
                    rs[tid] = rs[tid + off];
                    ri[tid] = ri[tid + off];
                    rslot[tid] = rslot[tid + off];
                }
            }
            __syncthreads();
        }
        if (tid == 0) {
            topS[it] = rs[0];
            topIdx[it] = ri[0];
            cs[rslot[0]] = -INFINITY;
        }
        __syncthreads();
    }

    // out[0 : 16*512] = memory_features[top_idx]; out[16*512 : +16] = top_scores
    for (int e = tid; e < TOPK * D; e += 256) {
        int r = e >> 9;       // D == 512
        int c = e & (D - 1);
        out[e] = mem[(long)topIdx[r] * D + c];
    }
    if (tid < TOPK) out[TOPK * D + tid] = topS[tid];
}

// ---------------- host launcher ----------------
extern "C" void kernel_launch(void* const* d_in, const int* in_sizes, int n_in,
                              void* d_out, int out_size, void* d_ws, size_t ws_size,
                              hipStream_t stream) {
    (void)in_sizes; (void)n_in; (void)out_size; (void)ws_size;
    const float* q   = (const float*)d_in[0];   // [512]
    const float* mem = (const float*)d_in[1];   // [500000, 512]
    float* out = (float*)d_out;                 // [16*512 + 16]

    // workspace layout (floats): [0]=qfac, [16..16+N) scores, then candS, then candI
    float* ws_f   = (float*)d_ws;
    float* qfac   = ws_f;
    float* scores = ws_f + 16;
    float* candS  = ws_f + 16 + NROWS;
    int*   candI  = (int*)(ws_f + 16 + NROWS + NCAND);

    qnorm_kernel<<<1, 256, 0, stream>>>(q, qfac);
    score_kernel<<<BLOCKS_A, TPB_A, 0, stream>>>(q, mem, qfac, scores);
    blocktop_kernel<<<NB2, TPB_B, 0, stream>>>(scores, candS, candI);
    merge_gather_kernel<<<1, 256, 0, stream>>>(candS, candI, mem, out);
}